// MatrixRouting_75625784148290
// MI455X (gfx1250) — compile-verified
//
#include <hip/hip_runtime.h>
#include <math.h>

// EM routing (matrix capsules) for MI455X / gfx1250, wave32.
// One 1024-thread block per batch element; lane == capsule c.
// 3 fused passes over votes (160MB, L2-resident after pass 1).

#define EPSF 1e-10f

typedef __attribute__((ext_vector_type(2))) float v2f;
typedef __attribute__((ext_vector_type(8))) float v8f;

#ifndef __has_builtin
#define __has_builtin(x) 0
#endif

#if __has_builtin(__builtin_amdgcn_wmma_f32_16x16x4_f32)
#define USE_WMMA_COSTSUM 1
#else
#define USE_WMMA_COSTSUM 0
#endif

__global__ __launch_bounds__(1024, 1)
void em_routing_kernel(const float* __restrict__ votes,
                       const float* __restrict__ beta_v,
                       const float* __restrict__ beta_a,
                       float* __restrict__ out)
{
  constexpr int N = 1152, C = 32, K = 16, H = 17;
  const int tid  = threadIdx.x;
  const int wave = tid >> 5;
  const int lane = tid & 31;          // lane == capsule index c in the main pass
  const int b    = blockIdx.x;

  __shared__ float  s_part[8][C][33];  // slot partials: [c][ S0 | S1[16] | S2[16] ]
  __shared__ float4 s_stat[C][K];      // x=mu  y=1/(2*sig2)  z=lnp const  w=log_sigma
  __shared__ float  s_a[C];
  __shared__ float  s_sumR[C];
  __shared__ float  s_costk[C];

  const float* vb = votes + (size_t)b * N * C * H;

  for (int iter = 0; iter < 3; ++iter) {
    // ---------------- accumulation pass over i (R-update fused for iter>0) ----
    float S0 = 0.f;
    float S1[K], S2[K];
#pragma unroll
    for (int k = 0; k < K; ++k) { S1[k] = 0.f; S2[k] = 0.f; }

    for (int i = wave; i < N; i += 32) {
      const float* row = vb + ((size_t)i * C + lane) * H;
      if (i + 32 < N)
        __builtin_prefetch(vb + ((size_t)(i + 32) * C + lane) * H, 0, 3);

      float v[K];
#pragma unroll
      for (int k = 0; k < K; ++k) v[k] = row[k];
      const float ain = row[16];

      float w;
      if (iter == 0) {
        w = ain * (1.0f / 32.0f);          // R = 1/C
      } else {
        // ln_p with previous iteration's stats; softmax-like normalize over c.
        float ps = 0.f;
#pragma unroll
        for (int k = 0; k < K; ++k) {
          float4 st = s_stat[lane][k];
          float d = v[k] - st.x;
          ps += __expf(fmaf(-d * d, st.y, st.z));
        }
        float ap  = s_a[lane] * ps;
        float tot = ap;
#pragma unroll
        for (int m = 16; m >= 1; m >>= 1) tot += __shfl_xor(tot, m, 32);
        float R = ap / (tot + EPSF) + EPSF;
        w = R * ain;
      }

      S0 += w;
#pragma unroll
      for (int k = 0; k < K; ++k) {
        S1[k] = fmaf(w, v[k], S1[k]);
        S2[k] = fmaf(w * v[k], v[k], S2[k]);
      }
    }

    __syncthreads();
    // -------- deterministic cross-wave reduction: 32 waves -> 8 LDS slots -----
    for (int r = 0; r < 4; ++r) {
      if ((wave >> 3) == r) {
        float* dst = &s_part[wave & 7][lane][0];
        if (r == 0) {
          dst[0] = S0;
#pragma unroll
          for (int k = 0; k < K; ++k) { dst[1 + k] = S1[k]; dst[17 + k] = S2[k]; }
        } else {
          dst[0] += S0;
#pragma unroll
          for (int k = 0; k < K; ++k) { dst[1 + k] += S1[k]; dst[17 + k] += S2[k]; }
        }
      }
      __syncthreads();
    }

    // ---------------- per-(c,k) Gaussian stats -------------------------------
    if (tid < C * K) {
      const int c = tid >> 4, k = tid & 15;
      float s0 = 0.f, s1 = 0.f, s2 = 0.f;
#pragma unroll
      for (int s = 0; s < 8; ++s) {
        s0 += s_part[s][c][0];
        s1 += s_part[s][c][1 + k];
        s2 += s_part[s][c][17 + k];
      }
      const float sumR = s0 + 1e-4f;
      const float mu   = s1 / sumR;
      // sum R*(V-mu)^2 = S2 - 2 mu S1 + mu^2 S0 ; Vm2 carries +EPS -> +EPS*S0
      const float varn = fmaf(mu * mu, s0, fmaf(-2.f * mu, s1, s2));
      const float sig2 = fmaf(EPSF, s0, varn) / sumR + EPSF;
      const float logsig = __logf(sqrtf(sig2) + EPSF);
      const float i2s  = 0.5f / sig2;
      float4 st;
      st.x = mu;
      st.y = i2s;
      st.z = fmaf(-EPSF, i2s, -logsig - 0.91893853320467274f); // -EPS*i2s - logsig - ln(2pi)/2
      st.w = logsig;
      s_stat[c][k] = st;
      if (k == 0) s_sumR[c] = sumR;
    }
    __syncthreads();

    // ---------------- k-reduction of log_sigma (cost term) -------------------
#if USE_WMMA_COSTSUM
    // Row-sum of the 32x16 log_sigma matrix on the matrix unit:
    // D = sum_q A_q(16x4) x Ones(4x16), so every D column holds the row sums.
    if (wave == 0) {
      const int half = lane >> 4, m = lane & 15;
      const v2f ones = {1.f, 1.f};
#pragma unroll
      for (int cb = 0; cb < 2; ++cb) {
        const int c = cb * 16 + m;
        v8f acc = {};
#pragma unroll
        for (int q = 0; q < 4; ++q) {
          v2f aT;                                   // A[m][k]: lanes0-15 K={0,1}, lanes16-31 K={2,3}
          aT[0] = s_stat[c][4 * q + 2 * half + 0].w;
          aT[1] = s_stat[c][4 * q + 2 * half + 1].w;
          acc = __builtin_amdgcn_wmma_f32_16x16x4_f32(false, aT, false, ones,
                                                      (short)0, acc, false, false);
        }
        // D column N=0: lane0 vgpr r -> M=r ; lane16 vgpr r -> M=r+8
        if (lane == 0) {
#pragma unroll
          for (int r = 0; r < 8; ++r) s_costk[cb * 16 + r] = acc[r];
        } else if (lane == 16) {
#pragma unroll
          for (int r = 0; r < 8; ++r) s_costk[cb * 16 + 8 + r] = acc[r];
        }
      }
    }
#else
    if (tid < C) {
      float s = 0.f;
#pragma unroll
      for (int k = 0; k < K; ++k) s += s_stat[tid][k].w;
      s_costk[tid] = s;
    }
#endif
    __syncthreads();

    // ---------------- activation -------------------------------------------
    if (tid < C) {
      const float lam = (iter == 0) ? 0.0005f : (iter == 1 ? 0.000975f : 0.00142625f);
      const float cost = s_sumR[tid] * fmaf(16.f, beta_v[tid], s_costk[tid]);
      const float x = lam * (beta_a[tid] - cost);
      s_a[tid] = 1.f / (1.f + __expf(-x));
    }
    __syncthreads();
  }

  // ---------------- output: concat(mu, a) -> (b, 32, 17) ---------------------
  if (tid < C * H) {
    const int c = tid / H, j = tid - c * H;
    out[(size_t)b * (C * H) + tid] = (j < 16) ? s_stat[c][j].x : s_a[c];
  }
}

extern "C" void kernel_launch(void* const* d_in, const int* in_sizes, int n_in,
                              void* d_out, int out_size, void* d_ws, size_t ws_size,
                              hipStream_t stream) {
  (void)in_sizes; (void)n_in; (void)out_size; (void)d_ws; (void)ws_size;
  const float* votes  = (const float*)d_in[0];
  const float* beta_v = (const float*)d_in[1];
  const float* beta_a = (const float*)d_in[2];
  float* out = (float*)d_out;
  em_routing_kernel<<<dim3(64), dim3(1024), 0, stream>>>(votes, beta_v, beta_a, out);
}